// GNNLayer_4818953306373
// MI455X (gfx1250) — compile-verified
//
#include <hip/hip_runtime.h>
#include <hip/hip_bf16.h>

typedef __attribute__((ext_vector_type(2))) float v2f;
typedef __attribute__((ext_vector_type(8))) float v8f;

constexpr int NN   = 50000;
constexpr int NE   = 800000;
constexpr int IND  = 256;
constexpr int HIDD = 256;
constexpr int OUTD = 64;
constexpr int NH   = 4;
constexpr int ATT  = 129;          // 2*OUTD + 1
constexpr int KPAD = 132;          // ATT padded to multiple of 4
constexpr int NPAD = 144;          // ATT padded to multiple of 16
constexpr int XS   = 260;          // LDS stride for 256-wide tiles (conflict-free)
constexpr int HSD  = 148;          // LDS stride for hid2 (144 cols)

// ---------------- init helpers ----------------
__global__ void k_zero(float* __restrict__ p, int n) {
  int i = blockIdx.x * 256 + threadIdx.x;
  if (i < n) p[i] = 0.f;
}

__global__ void k_initmax(unsigned* __restrict__ m) {
  if (threadIdx.x < NH) m[threadIdx.x] = 0x007FFFFFu;  // encode(-inf)
}

// Pack row-major W[h][K][N] into K-pair-interleaved Wp[h][K/2][N][2]
// so one 8-byte load per lane fetches the {k, k+1} B-fragment pair.
__global__ void k_packpairs(const float* __restrict__ W, float* __restrict__ Wp,
                            int K, int N) {
  int g = blockIdx.x * 256 + threadIdx.x;
  if (g >= NH * K * N) return;
  int h = g / (K * N);
  int r = (g / N) % K;
  int n = g % N;
  Wp[(((size_t)h * (K >> 1) + (r >> 1)) * N + n) * 2 + (r & 1)] = W[g];
}

// Zero-pad A1 [ATT x ATT] to [KPAD x NPAD] and K-pair interleave.
__global__ void k_packA1(const float* __restrict__ A1, float* __restrict__ A1p) {
  int g = blockIdx.x * 256 + threadIdx.x;
  if (g >= NH * KPAD * NPAD) return;
  int h = g / (KPAD * NPAD);
  int r = (g / NPAD) % KPAD;
  int n = g % NPAD;
  float v = (r < ATT && n < ATT) ? A1[(h * ATT + r) * ATT + n] : 0.f;
  A1p[(((size_t)h * (KPAD >> 1) + (r >> 1)) * NPAD + n) * 2 + (r & 1)] = v;
}

// ---------------- node MLP: feats = relu(x@W1+b1)@W2+b2 ----------------
__global__ __launch_bounds__(256) void k_node(const float* __restrict__ x,
                                              const float* __restrict__ W1p,
                                              const float* __restrict__ b1,
                                              const float* __restrict__ W2p,
                                              const float* __restrict__ b2,
                                              float* __restrict__ feats) {
  __shared__ float xs[16 * XS];
  __shared__ float hs[16 * XS];
  const int h = blockIdx.y;
  const int node0 = blockIdx.x * 16;
  const int tid = threadIdx.x, lane = tid & 31, wave = tid >> 5;

  for (int i = tid; i < 16 * IND; i += 256) {
    int r = i >> 8, c = i & 255;
    xs[r * XS + c] = x[(size_t)(node0 + r) * IND + c];
  }
  __syncthreads();

  const int m   = lane & 15;
  const int kb2 = (lane >> 4) << 1;   // 0 or 2 : per-lane K pair select
  const int mhi = (lane >> 4) << 3;   // 0 or 8 : D row offset

  // ---- layer 1: each wave does N-tiles {wave, wave+8} in one K loop ----
  {
    const int n0 = wave * 16 + (lane & 15);
    const int n1 = n0 + 128;
    v8f c0 = {}, c1 = {};
    const float* Wp = W1p + (size_t)h * IND * HIDD;   // packed [K/2][N][2]
    for (int k0 = 0; k0 < IND; k0 += 4) {
      int kk = k0 + kb2;                               // even
      v2f a; a.x = xs[m * XS + kk]; a.y = xs[m * XS + kk + 1];
      v2f b0 = *(const v2f*)(Wp + (((size_t)(kk >> 1)) * HIDD + n0) * 2);
      v2f b1 = *(const v2f*)(Wp + (((size_t)(kk >> 1)) * HIDD + n1) * 2);
      c0 = __builtin_amdgcn_wmma_f32_16x16x4_f32(false, a, false, b0, (short)0, c0, false, false);
      c1 = __builtin_amdgcn_wmma_f32_16x16x4_f32(false, a, false, b1, (short)0, c1, false, false);
    }
    float bias0 = b1[h * HIDD + n0];
    float bias1 = b1[h * HIDD + n1];
#pragma unroll
    for (int i = 0; i < 8; ++i) {
      float v0 = c0[i] + bias0, v1 = c1[i] + bias1;
      hs[(i + mhi) * XS + n0] = v0 > 0.f ? v0 : 0.f;
      hs[(i + mhi) * XS + n1] = v1 > 0.f ? v1 : 0.f;
    }
  }
  __syncthreads();

  // ---- layer 2: 4 N-tiles, K split across wave pairs, LDS reduce ----
  {
    const int t = wave & 3;          // tile
    const int kh = wave >> 2;        // K half: 0 or 1
    const int n = t * 16 + (lane & 15);
    v8f c = {};
    const float* Wp = W2p + (size_t)h * HIDD * OUTD;  // packed [K/2][N][2]
    const int kbeg = kh * 128, kend = kbeg + 128;
    for (int k0 = kbeg; k0 < kend; k0 += 4) {
      int kk = k0 + kb2;
      v2f a; a.x = hs[m * XS + kk]; a.y = hs[m * XS + kk + 1];
      v2f b = *(const v2f*)(Wp + (((size_t)(kk >> 1)) * OUTD + n) * 2);
      c = __builtin_amdgcn_wmma_f32_16x16x4_f32(false, a, false, b, (short)0, c, false, false);
    }
    __syncthreads();                 // xs no longer needed; reuse as partial buf
#pragma unroll
    for (int i = 0; i < 8; ++i) xs[(wave * 8 + i) * 32 + lane] = c[i];
  }
  __syncthreads();
  if (wave < 4) {
    const int n = wave * 16 + (lane & 15);
    float bias = b2[h * OUTD + n];
#pragma unroll
    for (int i = 0; i < 8; ++i) {
      float v = xs[(wave * 8 + i) * 32 + lane] + xs[((wave + 4) * 8 + i) * 32 + lane] + bias;
      feats[((size_t)h * NN + (node0 + i + mhi)) * OUTD + n] = v;
    }
  }
}

// ---------------- edge attention MLP (288 threads = 9 waves, 1 wave/N-tile) --
__global__ __launch_bounds__(288) void k_edge(const float* __restrict__ feats,
                                              const int* __restrict__ idx,
                                              const float* __restrict__ elem,
                                              const float* __restrict__ A1p,
                                              const float* __restrict__ ab1,
                                              const float* __restrict__ A2,
                                              const float* __restrict__ ab2,
                                              float* __restrict__ e1w,
                                              unsigned* __restrict__ maxenc) {
  __shared__ float xc[16 * KPAD];
  __shared__ float h2[16 * HSD];
  __shared__ float red[16 * 16];
  __shared__ float emax[16];
  const int h = blockIdx.y;
  const int e0 = blockIdx.x * 16;
  const int tid = threadIdx.x, lane = tid & 31, wave = tid >> 5;

  // gather x_cat = [feats[row] | feats[col] | elem | 0pad]
  for (int i = tid; i < 16 * KPAD; i += 288) {
    int r = i / KPAD, c = i % KPAD;
    int e = e0 + r;
    float v;
    if (c < OUTD)            v = feats[((size_t)h * NN + idx[e]) * OUTD + c];
    else if (c < 2 * OUTD)   v = feats[((size_t)h * NN + idx[NE + e]) * OUTD + (c - OUTD)];
    else if (c == 2 * OUTD)  v = elem[e];
    else                     v = 0.f;
    xc[r * KPAD + c] = v;
  }
  __syncthreads();

  const int m   = lane & 15;
  const int kb2 = (lane >> 4) << 1;
  const int mhi = (lane >> 4) << 3;

  // hid2 = relu(x_cat @ A1 + ab1) : 9 N-tiles, one per wave
  {
    const int n = wave * 16 + (lane & 15);    // wave 0..8 -> n < 144
    v8f c = {};
    const float* Ap = A1p + (size_t)h * KPAD * NPAD;  // packed [KPAD/2][NPAD][2]
    for (int k0 = 0; k0 < KPAD; k0 += 4) {
      int kk = k0 + kb2;
      v2f a; a.x = xc[m * KPAD + kk]; a.y = xc[m * KPAD + kk + 1];
      v2f b = *(const v2f*)(Ap + (((size_t)(kk >> 1)) * NPAD + n) * 2);
      c = __builtin_amdgcn_wmma_f32_16x16x4_f32(false, a, false, b, (short)0, c, false, false);
    }
    float bias = (n < ATT) ? ab1[h * ATT + n] : 0.f;
#pragma unroll
    for (int i = 0; i < 8; ++i) {
      float v = c[i] + bias;
      h2[(i + mhi) * HSD + n] = v > 0.f ? v : 0.f;
    }
  }
  __syncthreads();

  // e1 = hid2 @ A2 + ab2, 16 threads per edge (threads 0..255)
  if (tid < 256) {
    int r = tid >> 4, j = tid & 15;
    float s = 0.f;
    for (int jj = j; jj < ATT; jj += 16) s += h2[r * HSD + jj] * A2[h * ATT + jj];
    red[r * 16 + j] = s;
  }
  __syncthreads();
  if (tid < 256 && (tid & 15) == 0) {
    int r = tid >> 4;
    float s = 0.f;
#pragma unroll
    for (int j = 0; j < 16; ++j) s += red[r * 16 + j];
    s += ab2[h];
    s = (s > 0.f) ? s : 0.2f * s;  // leaky_relu
    e1w[(size_t)h * NE + e0 + r] = s;
    emax[r] = s;
  }
  __syncthreads();
  if (tid == 0) {
    float mx = emax[0];
#pragma unroll
    for (int r = 1; r < 16; ++r) mx = fmaxf(mx, emax[r]);
    unsigned u = __float_as_uint(mx);
    u = (u & 0x80000000u) ? ~u : (u | 0x80000000u);  // order-preserving encode
    atomicMax(&maxenc[h], u);
  }
}

// ---------------- weighted scatter (segment sums) ----------------
__global__ __launch_bounds__(256) void k_scatter(const float* __restrict__ feats,
                                                 const int* __restrict__ idx,
                                                 const float* __restrict__ e1w,
                                                 const unsigned* __restrict__ maxenc,
                                                 float* __restrict__ out,
                                                 float* __restrict__ rowsum) {
  const int tid = threadIdx.x;
  const int le = tid >> 6;            // 4 edges per block
  const int f  = tid & 63;
  const int e  = blockIdx.x * 4 + le;
  const int row = idx[e], col = idx[NE + e];
#pragma unroll
  for (int h = 0; h < NH; ++h) {
    unsigned u = maxenc[h];
    float mx = (u & 0x80000000u) ? __uint_as_float(u & 0x7FFFFFFFu) : __uint_as_float(~u);
    float w = __expf(e1w[(size_t)h * NE + e] - mx);
    float v = w * feats[((size_t)h * NN + col) * OUTD + f];
    atomicAdd(&out[(size_t)row * (NH * OUTD) + h * OUTD + f], v);
    if (f == 0) atomicAdd(&rowsum[h * NN + row], w);
  }
}

// ---------------- normalize ----------------
__global__ __launch_bounds__(256) void k_norm(float* __restrict__ out,
                                              const float* __restrict__ rowsum) {
  const int node = blockIdx.x, tid = threadIdx.x;
  const int h = tid >> 6;
  float rs = rowsum[h * NN + node] + 1e-10f;
  out[(size_t)node * (NH * OUTD) + tid] /= rs;
}

extern "C" void kernel_launch(void* const* d_in, const int* in_sizes, int n_in,
                              void* d_out, int out_size, void* d_ws, size_t ws_size,
                              hipStream_t stream) {
  const float* x    = (const float*)d_in[0];
  const int*   idx  = (const int*)d_in[1];
  const float* elem = (const float*)d_in[2];
  const float* W1   = (const float*)d_in[3];
  const float* b1   = (const float*)d_in[4];
  const float* W2   = (const float*)d_in[5];
  const float* b2   = (const float*)d_in[6];
  const float* A1   = (const float*)d_in[7];
  const float* ab1  = (const float*)d_in[8];
  const float* A2   = (const float*)d_in[9];
  const float* ab2  = (const float*)d_in[10];
  float* out = (float*)d_out;

  float* feats   = (float*)d_ws;                              // NH*NN*OUTD
  float* e1w     = feats + (size_t)NH * NN * OUTD;            // NH*NE
  float* rowsum  = e1w + (size_t)NH * NE;                     // NH*NN
  unsigned* maxenc = (unsigned*)(rowsum + (size_t)NH * NN);   // NH
  float* A1p     = (float*)(maxenc + NH);                     // NH*KPAD*NPAD
  float* W1p     = A1p + (size_t)NH * KPAD * NPAD;            // NH*IND*HIDD
  float* W2p     = W1p + (size_t)NH * IND * HIDD;             // NH*HIDD*OUTD

  k_zero<<<(NN * NH * OUTD + 255) / 256, 256, 0, stream>>>(out, NN * NH * OUTD);
  k_zero<<<(NH * NN + 255) / 256, 256, 0, stream>>>(rowsum, NH * NN);
  k_initmax<<<1, 32, 0, stream>>>(maxenc);
  k_packA1<<<(NH * KPAD * NPAD + 255) / 256, 256, 0, stream>>>(A1, A1p);
  k_packpairs<<<(NH * IND * HIDD + 255) / 256, 256, 0, stream>>>(W1, W1p, IND, HIDD);
  k_packpairs<<<(NH * HIDD * OUTD + 255) / 256, 256, 0, stream>>>(W2, W2p, HIDD, OUTD);

  dim3 gn(NN / 16, NH);
  k_node<<<gn, 256, 0, stream>>>(x, W1p, b1, W2p, b2, feats);

  dim3 ge(NE / 16, NH);
  k_edge<<<ge, 288, 0, stream>>>(feats, idx, elem, A1p, ab1, A2, ab2, e1w, maxenc);

  k_scatter<<<NE / 4, 256, 0, stream>>>(feats, idx, e1w, maxenc, out, rowsum);
  k_norm<<<NN, 256, 0, stream>>>(out, rowsum);
}